// LongShortTermTransformer_6914897346612
// MI455X (gfx1250) — compile-verified
//
// LongShortTermTransformer forward for MI455X (gfx1250), fp32.
// GEMMs: WMMA f32 16x16x4 with double-buffered Tensor-Data-Mover (TDM) tile
// staging into LDS (DMA for chunk k+1 overlaps WMMA on chunk k), dual
// accumulators to break the XDL accumulation dependency chain.
// Falls back to vectorized cooperative loads if the TDM builtin is absent.
//
// Assumptions (documented, no runtime check possible):
//  - d_in flattening = recursive insertion order of setup_inputs():
//      [0]=tgt (1600*4*256), [1]=curr_id_emb, then per layer (31 tensors):
//      n1_s,n1_b,n2_s,n2_b,n3_s,n3_b,
//      sa.wq,sa.bq,sa.wk,sa.bk,sa.wv,sa.bv,sa.wp,sa.bp,
//      wQ,bQ,wV,bV, lt.wp,lt.bp, st.rel_w,st.rel_b,st.wp,st.bp,
//      w1,b1,gn_s,gn_b,dw_w,w2,b2;  then final_s, final_b, h, w.
//  - WMMA f32 16x16x4 fragment layout per ISA 7.12.2.
//  - TDM D# layout per ISA 8.3/8.4; TENSORcnt completes in-order per wave,
//    so s_wait_tensorcnt(2) after issuing the next 2 tile DMAs guarantees the
//    current chunk's 2 DMAs have completed.

#include <hip/hip_runtime.h>
#include <math.h>

typedef float v2f __attribute__((ext_vector_type(2)));
typedef float v8f __attribute__((ext_vector_type(8)));
typedef unsigned int u32x4 __attribute__((ext_vector_type(4)));
typedef int i32x4 __attribute__((ext_vector_type(4)));
typedef int i32x8 __attribute__((ext_vector_type(8)));

#define R_ROWS   6400      // hw * n = 1600 * 4
#define DMODEL   256
#define DFF      1024
#define NHEAD    8
#define HD       32
#define LSEQ     1600
#define HDIM     40
#define WDIM     40
#define WIN      225       // 15x15 window
#define RELN     1800      // NHEAD * WIN
#define INV_SQRT_HD 0.17677669529663687f

// ---------------------------------------------------------------- reductions
__device__ __forceinline__ float waveSum(float v) {
#pragma unroll
  for (int off = 16; off; off >>= 1) v += __shfl_xor(v, off);
  return v;
}
__device__ __forceinline__ float waveMax(float v) {
#pragma unroll
  for (int off = 16; off; off >>= 1) v = fmaxf(v, __shfl_xor(v, off));
  return v;
}

// ---------------------------------------------------------------- TDM helpers
#define GEMM_BM 64
#define GEMM_BN 32
#define GEMM_KC 32
#define LDS_STRIDE (GEMM_KC + 4)   // 36 floats: 16B-aligned rows, conflict-free

#if __has_builtin(__builtin_amdgcn_tensor_load_to_lds)
#define HAVE_TDM 1
__device__ __forceinline__ void tdm_issue(u32x4 g0, i32x8 g1) {
  i32x4 z4 = {0, 0, 0, 0};
#if defined(__clang_major__) && __clang_major__ >= 23
  i32x8 z8 = {0, 0, 0, 0, 0, 0, 0, 0};
  __builtin_amdgcn_tensor_load_to_lds(g0, g1, z4, z4, z8, 0);
#else
  __builtin_amdgcn_tensor_load_to_lds(g0, g1, z4, z4, 0);
#endif
}
// 2D tile load, 4-byte elements, 32-element rows padded to 36 in LDS
// (pad_interval code 4 = 32 DWORDs, pad_amount code 3 = 4 DWORDs).
__device__ __forceinline__ void tdm_load_tile(
    const float* lds_dst, const float* gsrc, unsigned tensor_d0,
    unsigned tensor_d1, unsigned tile_d0, unsigned tile_d1, unsigned stride) {
  unsigned long long ga = (unsigned long long)(size_t)gsrc;
  u32x4 g0;
  g0[0] = 1u;                                           // count=1, no gather
  g0[1] = (unsigned)(size_t)lds_dst;                    // LDS byte address
  g0[2] = (unsigned)(ga & 0xFFFFFFFFu);                 // global_addr[31:0]
  g0[3] = (unsigned)((ga >> 32) & 0x01FFFFFFu) | (2u << 30);  // addr[56:32]|type=2
  i32x8 g1;
  g1[0] = (int)((2u << 16) | (1u << 20) | (4u << 22) | (3u << 25));
  g1[1] = (int)((tensor_d0 & 0xFFFFu) << 16);           // td0[15:0] @ bit48
  g1[2] = (int)((tensor_d0 >> 16) | ((tensor_d1 & 0xFFFFu) << 16));
  g1[3] = (int)((tensor_d1 >> 16) | (tile_d0 << 16));   // tile_dim0 @ bit112
  g1[4] = (int)(tile_d1 & 0xFFFFu);                     // tile_dim1, tile_dim2=0
  g1[5] = (int)stride;                                  // dim0_stride[31:0] (elems)
  g1[6] = 0;
  g1[7] = 0;
  tdm_issue(g0, g1);
}
#else
#define HAVE_TDM 0
#endif

// ---------------------------------------------------------------- WMMA GEMM
// C[M,N] = alpha * A[M,K] @ op(B) + bias[N] + resid[M,N]
// TRANSB=true : B is (N,K) row-major (weight layout, computes A@B^T)
// TRANSB=false: B is (K,N) row-major
// ALIGNED=true: M%64==0, N%32==0 (K handled by zero-fill) -> no guards,
//               double-buffered TDM staging.
template <bool TRANSB, bool ALIGNED>
__global__ __launch_bounds__(256) void gemm_wmma(
    const float* __restrict__ A, int lda, const float* __restrict__ B, int ldb,
    const float* __restrict__ bias, const float* __restrict__ resid,
    float* __restrict__ C, int ldc, int M, int N, int K, float alpha) {
  __shared__ float As[2][GEMM_BM * LDS_STRIDE];
  __shared__ float Bs[2][GEMM_BN * LDS_STRIDE];  // 32x36 both layouts
  const int tid = threadIdx.x;
  const int lane = tid & 31;
  const int wv = tid >> 5;        // 0..7
  const int tm = wv >> 1;         // 0..3  (M sub-tile)
  const int tn = wv & 1;          // 0..1  (N sub-tile)
  const int half = lane >> 4;
  const int l16 = lane & 15;
  const int rowBase = blockIdx.y * GEMM_BM;
  const int colBase = blockIdx.x * GEMM_BN;

#if HAVE_TDM
  // TDM issue for one K-chunk (A tile + B tile), wave-level DMA.
  auto issue_chunk = [&](int k0, int buf) {
    tdm_load_tile(As[buf], A + (size_t)rowBase * lda + k0,
                  (unsigned)(K - k0), (unsigned)(M - rowBase),
                  GEMM_KC, GEMM_BM, (unsigned)lda);
    if constexpr (TRANSB)  // n-major in LDS: Bs[n][k]
      tdm_load_tile(Bs[buf], B + (size_t)colBase * ldb + k0,
                    (unsigned)(K - k0), (unsigned)(N - colBase),
                    GEMM_KC, GEMM_BN, (unsigned)ldb);
    else                   // k-major in LDS: Bs[k][n]
      tdm_load_tile(Bs[buf], B + (size_t)k0 * ldb + colBase,
                    (unsigned)(N - colBase), (unsigned)(K - k0),
                    GEMM_BN, GEMM_KC, (unsigned)ldb);
  };
  if (ALIGNED && wv == 0) issue_chunk(0, 0);   // prologue: chunk 0 -> buf 0
#endif

  v8f acc0 = {}, acc1 = {};
  int cur = 0;
  for (int k0 = 0; k0 < K; k0 += GEMM_KC, cur ^= 1) {
    __syncthreads();  // (a) buf[cur^1] reads from previous iteration are done
    if constexpr (ALIGNED) {
#if HAVE_TDM
      if (wv == 0) {
        if (k0 + GEMM_KC < K) {
          issue_chunk(k0 + GEMM_KC, cur ^ 1);       // overlap next DMA
          __builtin_amdgcn_s_wait_tensorcnt(2);     // current chunk landed
        } else {
          __builtin_amdgcn_s_wait_tensorcnt(0);
        }
      }
#else
      // fallback: cooperative vectorized staging (b128 loads)
      {
        const float4* Ag = (const float4*)(A + (size_t)rowBase * lda + k0);
        int ldav = lda >> 2;
#pragma unroll
        for (int j = tid; j < GEMM_BM * (GEMM_KC / 4); j += 256) {
          int r = j >> 3, k4 = j & 7;
          *(float4*)&As[cur][r * LDS_STRIDE + k4 * 4] = Ag[(size_t)r * ldav + k4];
        }
        int ldbv = ldb >> 2;
        if constexpr (TRANSB) {
          const float4* Bg = (const float4*)(B + (size_t)colBase * ldb + k0);
#pragma unroll
          for (int j = tid; j < GEMM_BN * (GEMM_KC / 4); j += 256) {
            int n = j >> 3, k4 = j & 7;
            *(float4*)&Bs[cur][n * LDS_STRIDE + k4 * 4] = Bg[(size_t)n * ldbv + k4];
          }
        } else {
          const float4* Bg = (const float4*)(B + (size_t)k0 * ldb + colBase);
#pragma unroll
          for (int j = tid; j < GEMM_KC * (GEMM_BN / 4); j += 256) {
            int k = j >> 3, c4 = j & 7;
            *(float4*)&Bs[cur][k * LDS_STRIDE + c4 * 4] = Bg[(size_t)k * ldbv + c4];
          }
        }
      }
#endif
    } else {
      // guarded scalar staging (only used for N=1800 rel-bias GEMM)
#pragma unroll
      for (int i = tid; i < GEMM_BM * GEMM_KC; i += 256) {
        int r = i >> 5, kk = i & 31;
        int gr = rowBase + r, gk = k0 + kk;
        As[cur][r * LDS_STRIDE + kk] =
            (gr < M && gk < K) ? A[(size_t)gr * lda + gk] : 0.f;
      }
#pragma unroll
      for (int i = tid; i < GEMM_BN * GEMM_KC; i += 256) {
        int c = i >> 5, kk = i & 31;
        int gc = colBase + c, gk = k0 + kk;
        float v = 0.f;
        if (gc < N && gk < K)
          v = TRANSB ? B[(size_t)gc * ldb + gk] : B[(size_t)gk * ldb + gc];
        Bs[cur][TRANSB ? (c * LDS_STRIDE + kk) : (kk * LDS_STRIDE + c)] = v;
      }
    }
    __syncthreads();  // (b) buf[cur] ready for all waves
    const float* Arow = &As[cur][(tm * 16 + l16) * LDS_STRIDE + 2 * half];
    const float* Bbuf = Bs[cur];
    const float* Brow = &Bbuf[(tn * 16 + l16) * LDS_STRIDE + 2 * half];
    // Two independent accumulator chains: kk and kk+16.
#pragma unroll
    for (int kk = 0; kk < GEMM_KC / 2; kk += 4) {
      v2f a0, b0, a1, b1;
      a0.x = Arow[kk];
      a0.y = Arow[kk + 1];
      a1.x = Arow[kk + 16];
      a1.y = Arow[kk + 17];
      if constexpr (TRANSB) {
        b0.x = Brow[kk];
        b0.y = Brow[kk + 1];
        b1.x = Brow[kk + 16];
        b1.y = Brow[kk + 17];
      } else {
        b0.x = Bbuf[(kk + 2 * half) * LDS_STRIDE + tn * 16 + l16];
        b0.y = Bbuf[(kk + 2 * half + 1) * LDS_STRIDE + tn * 16 + l16];
        b1.x = Bbuf[(kk + 16 + 2 * half) * LDS_STRIDE + tn * 16 + l16];
        b1.y = Bbuf[(kk + 16 + 2 * half + 1) * LDS_STRIDE + tn * 16 + l16];
      }
      acc0 = __builtin_amdgcn_wmma_f32_16x16x4_f32(
          false, a0, false, b0, (short)0, acc0, false, false);
      acc1 = __builtin_amdgcn_wmma_f32_16x16x4_f32(
          false, a1, false, b1, (short)0, acc1, false, false);
    }
  }
#pragma unroll
  for (int i = 0; i < 8; i++) {
    int gr = rowBase + tm * 16 + i + 8 * half;
    int gc = colBase + tn * 16 + l16;
    if (ALIGNED || (gr < M && gc < N)) {
      float v = alpha * (acc0[i] + acc1[i]);
      if (bias)  v += bias[gc];
      if (resid) v += resid[(size_t)gr * ldc + gc];
      C[(size_t)gr * ldc + gc] = v;
    }
  }
}

// ---------------------------------------------------------------- LayerNorm (256 cols, 1 wave/row)
__global__ __launch_bounds__(256) void ln256(
    const float* __restrict__ in, const float* __restrict__ s,
    const float* __restrict__ b, float* __restrict__ out, int rows) {
  int wv = threadIdx.x >> 5, lane = threadIdx.x & 31;
  int row = blockIdx.x * 8 + wv;
  if (row >= rows) return;
  const float* p = in + (size_t)row * DMODEL;
  float v[8], sum = 0.f, sq = 0.f;
#pragma unroll
  for (int j = 0; j < 8; j++) {
    v[j] = p[lane + 32 * j];
    sum += v[j];
    sq += v[j] * v[j];
  }
  sum = waveSum(sum);
  sq = waveSum(sq);
  float mu = sum * (1.f / DMODEL);
  float var = sq * (1.f / DMODEL) - mu * mu;
  float rstd = rsqrtf(var + 1e-5f);
  float* q = out + (size_t)row * DMODEL;
#pragma unroll
  for (int j = 0; j < 8; j++) {
    int c = lane + 32 * j;
    q[c] = (v[j] - mu) * rstd * s[c] + b[c];
  }
}

// ---------------------------------------------------------------- row softmax
__global__ __launch_bounds__(256) void softmax_rows(float* __restrict__ S, int cols) {
  float* p = S + (size_t)blockIdx.x * cols;
  int tid = threadIdx.x, lane = tid & 31, wv = tid >> 5;
  __shared__ float red[8];
  float m = -3.0e38f;
  for (int c = tid; c < cols; c += 256) m = fmaxf(m, p[c]);
  m = waveMax(m);
  if (lane == 0) red[wv] = m;
  __syncthreads();
  m = red[0];
#pragma unroll
  for (int i = 1; i < 8; i++) m = fmaxf(m, red[i]);
  __syncthreads();
  float s = 0.f;
  for (int c = tid; c < cols; c += 256) {
    float e = __expf(p[c] - m);
    p[c] = e;
    s += e;
  }
  s = waveSum(s);
  if (lane == 0) red[wv] = s;
  __syncthreads();
  s = 0.f;
#pragma unroll
  for (int i = 0; i < 8; i++) s += red[i];
  float inv = 1.f / s;
  for (int c = tid; c < cols; c += 256) p[c] *= inv;
}

// ---------------------------------------------------------------- elementwise add
__global__ __launch_bounds__(256) void add_elems(
    const float* __restrict__ a, const float* __restrict__ b,
    float* __restrict__ o, int n) {
  int i = blockIdx.x * 256 + threadIdx.x;
  if (i < n) o[i] = a[i] + b[i];
}

// ---------------------------------------------------------------- local windowed attention
// One wave per (spatial pos, batch n, head); lane = head dim (hd==32==wave32).
__global__ __launch_bounds__(256) void local_attn(
    const float* __restrict__ Q, const float* __restrict__ V,
    const float* __restrict__ rel, float* __restrict__ O) {
  __shared__ float qk[8][WIN];
  int wv = threadIdx.x >> 5, lane = threadIdx.x & 31;
  int pair = blockIdx.x * 8 + wv;          // 1600*4*8 = 51200 waves
  int h_ = pair & 7;
  int rn = pair >> 3;                      // row = pos*4 + n
  int n = rn & 3;
  int pos = rn >> 2;
  int y = pos / WDIM, x = pos % WDIM;
  float qd = Q[(size_t)rn * DMODEL + h_ * HD + lane] * INV_SQRT_HD;
  const float* relrow = rel + (size_t)rn * RELN + h_ * WIN;
  float mx = -3.0e38f;
  for (int o = 0; o < WIN; o++) {
    int ky = y + o / 15 - 7, kx = x + o % 15 - 7;
    float sv = -1e9f;
    if (ky >= 0 && ky < HDIM && kx >= 0 && kx < WDIM) {
      int kr = ((ky * WDIM + kx) << 2) + n;
      float t = qd * Q[(size_t)kr * DMODEL + h_ * HD + lane];
      t = waveSum(t);
      sv = t + relrow[o];
    }
    if (lane == 0) qk[wv][o] = sv;
    mx = fmaxf(mx, sv);
  }
  float part = 0.f;
  for (int o = lane; o < WIN; o += 32) {
    float e = __expf(qk[wv][o] - mx);
    qk[wv][o] = e;
    part += e;
  }
  part = waveSum(part);
  float inv = 1.f / part;
  float acc = 0.f;
  for (int o = 0; o < WIN; o++) {
    int ky = y + o / 15 - 7, kx = x + o % 15 - 7;
    if (ky >= 0 && ky < HDIM && kx >= 0 && kx < WDIM) {
      int kr = ((ky * WDIM + kx) << 2) + n;
      acc += qk[wv][o] * V[(size_t)kr * DMODEL + h_ * HD + lane];
    }
  }
  O[(size_t)rn * DMODEL + h_ * HD + lane] = acc * inv;
}

// ---------------------------------------------------------------- GroupNorm stats (per (n,g), 128 blocks)
__global__ __launch_bounds__(256) void gn_stats(
    const float* __restrict__ F, float* __restrict__ mean, float* __restrict__ rstd) {
  int b = blockIdx.x;               // b = n*32 + g
  int n = b >> 5, g = b & 31;
  int tid = threadIdx.x, lane = tid & 31, wv = tid >> 5;
  __shared__ float rs[8], rq[8];
  float s = 0.f, q = 0.f;
  for (int i = tid; i < LSEQ * 32; i += 256) {
    int pos = i >> 5;
    int c = (g << 5) + (i & 31);
    float v = F[((size_t)(pos << 2) + n) * DFF + c];
    s += v;
    q += v * v;
  }
  s = waveSum(s);
  q = waveSum(q);
  if (lane == 0) { rs[wv] = s; rq[wv] = q; }
  __syncthreads();
  if (tid == 0) {
    float ts = 0.f, tq = 0.f;
#pragma unroll
    for (int i = 0; i < 8; i++) { ts += rs[i]; tq += rq[i]; }
    const float invc = 1.f / (float)(LSEQ * 32);
    float mu = ts * invc;
    float var = tq * invc - mu * mu;
    mean[b] = mu;
    rstd[b] = rsqrtf(var + 1e-5f);
  }
}

// ---------------------------------------------------------------- GN apply + exact gelu
__global__ __launch_bounds__(256) void gn_apply_gelu(
    const float* __restrict__ F, const float* __restrict__ mean,
    const float* __restrict__ rstd, const float* __restrict__ gs,
    const float* __restrict__ gb, float* __restrict__ out) {
  int idx = blockIdx.x * 256 + threadIdx.x;
  if (idx >= R_ROWS * DFF) return;
  int c = idx & (DFF - 1);
  int r = idx >> 10;
  int n = r & 3;
  int p = (n << 5) | (c >> 5);
  float t = (F[idx] - mean[p]) * rstd[p] * gs[c] + gb[c];
  out[idx] = 0.5f * t * (1.f + erff(t * 0.70710678118654752f));
}

// ---------------------------------------------------------------- 5x5 depthwise conv (pad 2, no bias)
__global__ __launch_bounds__(256) void dwconv5x5(
    const float* __restrict__ in, const float* __restrict__ wgt,
    float* __restrict__ out) {
  int idx = blockIdx.x * 256 + threadIdx.x;
  if (idx >= R_ROWS * DFF) return;
  int c = idx & (DFF - 1);
  int r = idx >> 10;
  int n = r & 3;
  int pos = r >> 2;
  int y = pos / WDIM, x = pos % WDIM;
  float acc = 0.f;
#pragma unroll
  for (int ky = 0; ky < 5; ky++) {
    int yy = y + ky - 2;
    if (yy < 0 || yy >= HDIM) continue;
#pragma unroll
    for (int kx = 0; kx < 5; kx++) {
      int xx = x + kx - 2;
      if (xx < 0 || xx >= WDIM) continue;
      int rr = ((yy * WDIM + xx) << 2) + n;
      acc += wgt[c * 25 + ky * 5 + kx] * in[(size_t)rr * DFF + c];
    }
  }
  out[idx] = acc;
}

// ================================================================ host
extern "C" void kernel_launch(void* const* d_in, const int* in_sizes, int n_in,
                              void* d_out, int out_size, void* d_ws, size_t ws_size,
                              hipStream_t stream) {
  (void)in_sizes; (void)n_in; (void)out_size;
  const float* tgt = (const float*)d_in[0];
  const float* idemb = (const float*)d_in[1];
  auto P = [&](int i) -> const float* { return (const float*)d_in[i]; };

  // workspace arena (256B-aligned slabs)
  size_t cur = 0;
  auto alloc = [&](size_t elems) -> float* {
    float* p = (float*)((char*)d_ws + cur);
    cur += elems * sizeof(float);
    cur = (cur + 255) & ~(size_t)255;
    return p;
  };
  float* x   = alloc((size_t)R_ROWS * DMODEL);   // residual stream
  float* t0  = alloc((size_t)R_ROWS * DMODEL);   // LN output
  float* t1  = alloc((size_t)R_ROWS * DMODEL);   // _tgt + curr_id_emb
  float* Qb  = alloc((size_t)R_ROWS * DMODEL);
  float* Kb  = alloc((size_t)R_ROWS * DMODEL);
  float* Vb  = alloc((size_t)R_ROWS * DMODEL);
  float* Ob  = alloc((size_t)R_ROWS * DMODEL);   // global attn out
  float* O2b = alloc((size_t)R_ROWS * DMODEL);   // local attn out
  float* S   = alloc((size_t)LSEQ * LSEQ);       // score slab (reused per pair)
  float* F   = alloc((size_t)R_ROWS * DFF);
  float* F2  = alloc((size_t)R_ROWS * DFF);
  float* F3  = alloc((size_t)R_ROWS * DFF);
  float* gnm = alloc(128);
  float* gnr = alloc(128);
  float* rel = alloc((size_t)R_ROWS * RELN);
  if (cur > ws_size) return;  // insufficient scratch

  hipMemcpyAsync(x, tgt, (size_t)R_ROWS * DMODEL * sizeof(float),
                 hipMemcpyDeviceToDevice, stream);

  auto gemm_nt = [&](const float* A, int lda, const float* B, int ldb,
                     const float* bias, const float* resid, float* C, int ldc,
                     int M, int N, int K, float alpha) {
    dim3 g((N + GEMM_BN - 1) / GEMM_BN, (M + GEMM_BM - 1) / GEMM_BM);
    gemm_wmma<true, true><<<g, 256, 0, stream>>>(A, lda, B, ldb, bias, resid, C, ldc, M, N, K, alpha);
  };
  auto gemm_nn = [&](const float* A, int lda, const float* B, int ldb,
                     const float* bias, const float* resid, float* C, int ldc,
                     int M, int N, int K, float alpha) {
    dim3 g((N + GEMM_BN - 1) / GEMM_BN, (M + GEMM_BM - 1) / GEMM_BM);
    gemm_wmma<false, true><<<g, 256, 0, stream>>>(A, lda, B, ldb, bias, resid, C, ldc, M, N, K, alpha);
  };
  auto gemm_nt_guard = [&](const float* A, int lda, const float* B, int ldb,
                           const float* bias, const float* resid, float* C, int ldc,
                           int M, int N, int K, float alpha) {
    dim3 g((N + GEMM_BN - 1) / GEMM_BN, (M + GEMM_BM - 1) / GEMM_BM);
    gemm_wmma<true, false><<<g, 256, 0, stream>>>(A, lda, B, ldb, bias, resid, C, ldc, M, N, K, alpha);
  };

  for (int l = 0; l < 2; l++) {
    int base = 2 + l * 31;
    const float *n1s = P(base + 0), *n1b = P(base + 1);
    const float *n2s = P(base + 2), *n2b = P(base + 3);
    const float *n3s = P(base + 4), *n3b = P(base + 5);
    const float *wq = P(base + 6), *bq = P(base + 7);
    const float *wk = P(base + 8), *bk = P(base + 9);
    const float *wvv = P(base + 10), *bv = P(base + 11);
    const float *wp = P(base + 12), *bp = P(base + 13);
    const float *wQ = P(base + 14), *bQ = P(base + 15);
    const float *wV = P(base + 16), *bV = P(base + 17);
    const float *ltwp = P(base + 18), *ltbp = P(base + 19);
    const float *relw = P(base + 20), *relb = P(base + 21);
    const float *stwp = P(base + 22), *stbp = P(base + 23);
    const float *w1 = P(base + 24), *b1 = P(base + 25);
    const float *gns = P(base + 26), *gnb = P(base + 27);
    const float *dww = P(base + 28);
    const float *w2 = P(base + 29), *b2 = P(base + 30);

    // ---- self attention
    ln256<<<R_ROWS / 8, 256, 0, stream>>>(x, n1s, n1b, t0, R_ROWS);
    gemm_nt(t0, DMODEL, wq, DMODEL, bq, nullptr, Qb, DMODEL, R_ROWS, DMODEL, DMODEL, 1.f);
    gemm_nt(t0, DMODEL, wk, DMODEL, bk, nullptr, Kb, DMODEL, R_ROWS, DMODEL, DMODEL, 1.f);
    gemm_nt(t0, DMODEL, wvv, DMODEL, bv, nullptr, Vb, DMODEL, R_ROWS, DMODEL, DMODEL, 1.f);
    for (int p = 0; p < 32; p++) {           // p = n*8 + h, column offset p*32
      int off = p * HD;
      gemm_nt(Qb + off, NHEAD * HD * 4, Kb + off, NHEAD * HD * 4, nullptr, nullptr,
              S, LSEQ, LSEQ, LSEQ, HD, INV_SQRT_HD);
      softmax_rows<<<LSEQ, 256, 0, stream>>>(S, LSEQ);
      gemm_nn(S, LSEQ, Vb + off, NHEAD * HD * 4, nullptr, nullptr,
              Ob + off, NHEAD * HD * 4, LSEQ, HD, LSEQ, 1.f);
    }
    gemm_nt(Ob, DMODEL, wp, DMODEL, bp, x, x, DMODEL, R_ROWS, DMODEL, DMODEL, 1.f);

    // ---- shared Q/V projections
    ln256<<<R_ROWS / 8, 256, 0, stream>>>(x, n2s, n2b, t0, R_ROWS);
    gemm_nt(t0, DMODEL, wQ, DMODEL, bQ, nullptr, Qb, DMODEL, R_ROWS, DMODEL, DMODEL, 1.f);
    add_elems<<<(R_ROWS * DMODEL) / 256, 256, 0, stream>>>(t0, idemb, t1, R_ROWS * DMODEL);
    gemm_nt(t1, DMODEL, wV, DMODEL, bV, nullptr, Vb, DMODEL, R_ROWS, DMODEL, DMODEL, 1.f);

    // ---- long-term (global) attention, K = Q
    for (int p = 0; p < 32; p++) {
      int off = p * HD;
      gemm_nt(Qb + off, NHEAD * HD * 4, Qb + off, NHEAD * HD * 4, nullptr, nullptr,
              S, LSEQ, LSEQ, LSEQ, HD, INV_SQRT_HD);
      softmax_rows<<<LSEQ, 256, 0, stream>>>(S, LSEQ);
      gemm_nn(S, LSEQ, Vb + off, NHEAD * HD * 4, nullptr, nullptr,
              Ob + off, NHEAD * HD * 4, LSEQ, HD, LSEQ, 1.f);
    }
    // ---- short-term (local) attention: rel bias GEMM then windowed attention
    gemm_nt_guard(Qb, DMODEL, relw, DMODEL, relb, nullptr, rel, RELN, R_ROWS, RELN, DMODEL, 1.f);
    local_attn<<<R_ROWS * NHEAD / 8, 256, 0, stream>>>(Qb, Vb, rel, O2b);
    // residual adds via projection epilogues
    gemm_nt(Ob, DMODEL, ltwp, DMODEL, ltbp, x, x, DMODEL, R_ROWS, DMODEL, DMODEL, 1.f);
    gemm_nt(O2b, DMODEL, stwp, DMODEL, stbp, x, x, DMODEL, R_ROWS, DMODEL, DMODEL, 1.f);

    // ---- FFN with GN + gelu + depthwise conv
    ln256<<<R_ROWS / 8, 256, 0, stream>>>(x, n3s, n3b, t0, R_ROWS);
    gemm_nt(t0, DMODEL, w1, DMODEL, b1, nullptr, F, DFF, R_ROWS, DFF, DMODEL, 1.f);
    gn_stats<<<128, 256, 0, stream>>>(F, gnm, gnr);
    gn_apply_gelu<<<(R_ROWS * DFF) / 256, 256, 0, stream>>>(F, gnm, gnr, gns, gnb, F2);
    dwconv5x5<<<(R_ROWS * DFF) / 256, 256, 0, stream>>>(F2, dww, F3);
    gemm_nt(F3, DFF, w2, DFF, b2, x, x, DMODEL, R_ROWS, DMODEL, DFF, 1.f);
  }

  // final decoder LayerNorm -> d_out
  ln256<<<R_ROWS / 8, 256, 0, stream>>>(x, P(64), P(65), (float*)d_out, R_ROWS);
}